// AVQSeg_1529008357400
// MI455X (gfx1250) — compile-verified
//
#include <hip/hip_runtime.h>
#include <cfloat>

typedef __attribute__((ext_vector_type(2))) float v2f;
typedef __attribute__((ext_vector_type(8))) float v8f;

#define NPIX   32768          // B*H*W = 8*64*64
#define KCODES 8192
#define CDIM   64
#define HWSZ   4096           // H*W
#define KSPLIT 4
#define KPB    (KCODES / KSPLIT)   // 2048 codes per block
#define PIXBLK 128                 // pixel-tile blocks (32768 / 256)

// ---------------------------------------------------------------------------
// Kernel 1: per-code squared norms ||e_k||^2 -> d_ws
// ---------------------------------------------------------------------------
__global__ __launch_bounds__(256) void enorm_kernel(const float* __restrict__ emb,
                                                    float* __restrict__ enorm) {
    int k = blockIdx.x * 256 + threadIdx.x;
    const float4* row = (const float4*)(emb + (size_t)k * CDIM);
    float s = 0.f;
#pragma unroll
    for (int i = 0; i < 16; ++i) {
        float4 v = row[i];
        s = fmaf(v.x, v.x, s); s = fmaf(v.y, v.y, s);
        s = fmaf(v.z, v.z, s); s = fmaf(v.w, v.w, s);
    }
    enorm[k] = s;
}

// ---------------------------------------------------------------------------
// Kernel 2: seed packed (dist,idx) keys with +inf (must re-init every call).
// ---------------------------------------------------------------------------
__global__ __launch_bounds__(256) void initkey_kernel(unsigned long long* __restrict__ key) {
    key[blockIdx.x * 256 + threadIdx.x] = ~0ull;
}

// ---------------------------------------------------------------------------
// Kernel 3: fused partial GEMM + argmin. Grid = (128 pixel blocks, 4 K-splits).
// Each wave owns 32 pixels (two 16-pixel B tiles) so every A fragment (16
// codes x 4 channels) feeds two WMMAs: 32 v_wmma per 16 A-loads, 4
// independent accumulation chains. Partial winners merge across K-splits via
// global_atomic_min_u64 on a monotone (float-key<<32 | index) packing.
// ---------------------------------------------------------------------------
__global__ __launch_bounds__(256) void vq_argmin_kernel(const float* __restrict__ x,
                                                        const float* __restrict__ emb,
                                                        const float* __restrict__ enorm,
                                                        unsigned long long* __restrict__ key) {
    __shared__ float snorm[KPB];                     // 8 KB norm slice for this K-split

    const int kbase = blockIdx.y * KPB;

    // Stage this block's norm slice into LDS (each thread: 8 floats).
    {
        const float4* src = (const float4*)(enorm + kbase) + threadIdx.x * 2;
        float4 a = src[0], b = src[1];
        float4* dst = (float4*)snorm + threadIdx.x * 2;
        dst[0] = a; dst[1] = b;
    }
    __syncthreads();

    const int lane   = threadIdx.x & 31;
    const int wave   = threadIdx.x >> 5;
    const int l16    = lane & 15;
    const int laneHi = lane >> 4;
    const int h2     = laneHi * 2;

    const int pbase = (blockIdx.x * 8 + wave) * 32;  // 32 pixels per wave

    // Load both B tiles (pixel columns) into VGPR fragments, kept for the loop.
    v2f bx0[16], bx1[16];
#pragma unroll
    for (int t = 0; t < 2; ++t) {
        const int p  = pbase + t * 16 + l16;
        const int b  = p >> 12;
        const int hw = p & (HWSZ - 1);
        const float* xp = x + (size_t)b * CDIM * HWSZ + hw;
#pragma unroll
        for (int s = 0; s < 16; ++s) {
            const int c0 = 4 * s + h2;
            v2f v;
            v.x = xp[(size_t)c0 * HWSZ];
            v.y = xp[(size_t)(c0 + 1) * HWSZ];
            if (t == 0) bx0[s] = v; else bx1[s] = v;
        }
    }

    float minv[2][8];
    int   mini[2][8];
#pragma unroll
    for (int r = 0; r < 8; ++r) {
        minv[0][r] = FLT_MAX; mini[0][r] = 0;
        minv[1][r] = FLT_MAX; mini[1][r] = 0;
    }

    const float* arow = emb + (size_t)(kbase + l16) * CDIM + h2;

    for (int cc = 0; cc < KPB; cc += 16) {
        const float* erow = arow + (size_t)cc * CDIM;

        if (cc + 16 < KPB)
            __builtin_prefetch(erow + 16 * CDIM, 0, 0);

        v8f accA0 = {}, accA1 = {}, accB0 = {}, accB1 = {};
#pragma unroll
        for (int s = 0; s < 16; s += 2) {            // K=64, 4 chains of 8 WMMAs
            v2f a0, a1;
            a0.x = erow[4 * s];     a0.y = erow[4 * s + 1];
            a1.x = erow[4 * s + 4]; a1.y = erow[4 * s + 5];
            accA0 = __builtin_amdgcn_wmma_f32_16x16x4_f32(false, a0, false, bx0[s],
                                                          (short)0, accA0, false, false);
            accB0 = __builtin_amdgcn_wmma_f32_16x16x4_f32(false, a0, false, bx1[s],
                                                          (short)0, accB0, false, false);
            accA1 = __builtin_amdgcn_wmma_f32_16x16x4_f32(false, a1, false, bx0[s + 1],
                                                          (short)0, accA1, false, false);
            accB1 = __builtin_amdgcn_wmma_f32_16x16x4_f32(false, a1, false, bx1[s + 1],
                                                          (short)0, accB1, false, false);
        }

        // Norms for this lane's 8 code rows, broadcast from LDS.
        const int clocal = cc + laneHi * 8;
        const float4* np = (const float4*)(snorm + clocal);
        float4 n0 = np[0], n1 = np[1];
        float nv[8] = {n0.x, n0.y, n0.z, n0.w, n1.x, n1.y, n1.z, n1.w};
        const int cbase = kbase + clocal;

#pragma unroll
        for (int r = 0; r < 8; ++r) {
            float vA = fmaf(-2.0f, accA0[r] + accA1[r], nv[r]);
            float vB = fmaf(-2.0f, accB0[r] + accB1[r], nv[r]);
            bool tA = vA < minv[0][r];
            bool tB = vB < minv[1][r];
            mini[0][r] = tA ? (cbase + r) : mini[0][r];
            minv[0][r] = tA ? vA : minv[0][r];
            mini[1][r] = tB ? (cbase + r) : mini[1][r];
            minv[1][r] = tB ? vB : minv[1][r];
        }
    }

    // Reduce each pixel tile: r-scan (increasing code order), cross-half merge,
    // then global atomic-min on the packed monotone key.
#pragma unroll
    for (int t = 0; t < 2; ++t) {
        float bv = minv[t][0];
        int   bi = mini[t][0];
#pragma unroll
        for (int r = 1; r < 8; ++r) {
            bool take = (minv[t][r] < bv) || (minv[t][r] == bv && mini[t][r] < bi);
            bv = take ? minv[t][r] : bv;
            bi = take ? mini[t][r] : bi;
        }
        float ov = __shfl_xor(bv, 16, 32);
        int   oi = __shfl_xor(bi, 16, 32);
        if (ov < bv || (ov == bv && oi < bi)) { bv = ov; bi = oi; }

        if (laneHi == 0) {
            unsigned int u = __float_as_uint(bv);
            u = (u & 0x80000000u) ? ~u : (u | 0x80000000u);   // order-preserving map
            unsigned long long k64 =
                ((unsigned long long)u << 32) | (unsigned int)bi;
            atomicMin(&key[pbase + t * 16 + l16], k64);
        }
    }
}

// ---------------------------------------------------------------------------
// Kernel 4: gather winning codes and emit the three outputs.
//   out0 = x_q in BCHW, out1 = x_q in BHWC, out2 = x in BHWC
// ---------------------------------------------------------------------------
__global__ __launch_bounds__(256) void writeout_kernel(const float* __restrict__ x,
                                                       const float* __restrict__ emb,
                                                       const unsigned long long* __restrict__ key,
                                                       float* __restrict__ out0,
                                                       float* __restrict__ out1,
                                                       float* __restrict__ out2) {
    int t  = blockIdx.x * 256 + threadIdx.x;
    int n  = t >> 4;
    int c4 = (t & 15) * 4;
    int b  = n >> 12;
    int hw = n & (HWSZ - 1);

    int k = (int)(key[n] & 0xFFFFFFFFull);
    float4 e4 = *(const float4*)(emb + (size_t)k * CDIM + c4);

    *(float4*)(out1 + (size_t)n * CDIM + c4) = e4;

    const float* xp = x + ((size_t)b * CDIM + c4) * HWSZ + hw;
    float4 xv;
    xv.x = xp[0];
    xv.y = xp[HWSZ];
    xv.z = xp[2 * HWSZ];
    xv.w = xp[3 * HWSZ];
    *(float4*)(out2 + (size_t)n * CDIM + c4) = xv;

    float* o0 = out0 + ((size_t)b * CDIM + c4) * HWSZ + hw;
    o0[0]        = e4.x;
    o0[HWSZ]     = e4.y;
    o0[2 * HWSZ] = e4.z;
    o0[3 * HWSZ] = e4.w;
}

// ---------------------------------------------------------------------------
extern "C" void kernel_launch(void* const* d_in, const int* in_sizes, int n_in,
                              void* d_out, int out_size, void* d_ws, size_t ws_size,
                              hipStream_t stream) {
    const float* x   = (const float*)d_in[0];   // (8, 64, 64, 64) f32
    const float* emb = (const float*)d_in[1];   // (8192, 64) f32

    float*              enorm = (float*)d_ws;                                   // 32 KB
    unsigned long long* key   = (unsigned long long*)((char*)d_ws + 32768);     // 256 KB

    float* out0 = (float*)d_out;
    float* out1 = out0 + (size_t)NPIX * CDIM;
    float* out2 = out1 + (size_t)NPIX * CDIM;

    enorm_kernel<<<KCODES / 256, 256, 0, stream>>>(emb, enorm);
    initkey_kernel<<<NPIX / 256, 256, 0, stream>>>(key);
    vq_argmin_kernel<<<dim3(PIXBLK, KSPLIT), 256, 0, stream>>>(x, emb, enorm, key);
    writeout_kernel<<<(NPIX * 16) / 256, 256, 0, stream>>>(x, emb, key, out0, out1, out2);
}